// FeatureRotation_64338610094830
// MI455X (gfx1250) — compile-verified
//
#include <hip/hip_runtime.h>
#include <hip/hip_bf16.h>

// FeatureRotation: out[n,j,p,q] = sum_i W[i,j] * x[n,i,p,q]
//   == Out[n] = W^T (256x256) @ X[n] (256x3136), fp32.
// Roofline: 26.3 GFLOP vs 411 MB HBM traffic -> memory floor ~17.6 us @23.3TB/s.
// fp32 WMMA (v_wmma_f32_16x16x4_f32) keeps exact precision; W (256KB) stays L2-hot.

typedef __attribute__((ext_vector_type(2))) float v2f;
typedef __attribute__((ext_vector_type(8))) float v8f;

#define C_CH 256   // channels (GEMM K and M)
#define HW   3136  // 56*56 spatial (GEMM N)
#define BN   64    // spatial columns per workgroup
#define BK   32    // K block staged in LDS
#define NPAIR (BK / 2)   // 16 interleaved k-pairs
#define PSTR  160        // floats per k-pair row: 64*2 + 32 pad
                         // 160 mod 64 == 32 -> lanes 16-31 hit the other bank half

// ---- one-time W transpose into pair-interleaved layout:
//      Wp[kp*512 + m*2 + par] = W[(2*kp+par)*256 + m]
//      -> A fragment {W[k][m], W[k+1][m]} is one contiguous, coalesced b64.
__global__ __launch_bounds__(256)
void transpose_w_pairs(const float* __restrict__ Wm, float* __restrict__ Wp) {
    const int o   = blockIdx.x * 256 + threadIdx.x;  // 65536 outputs, coalesced
    const int kp  = o >> 9;
    const int rem = o & 511;
    const int m   = rem >> 1;
    const int par = rem & 1;
    Wp[o] = Wm[(2 * kp + par) * C_CH + m];
}

template <bool PRET>
__global__ __launch_bounds__(256, 2)
void feature_rotation_wmma_f32(const float* __restrict__ x,
                               const float* __restrict__ Wsrc,  // Wp if PRET, else W
                               float* __restrict__ out) {
    __shared__ float Bs[NPAIR * PSTR];

    const int tid  = threadIdx.x;
    const int lane = tid & 31;   // wave32
    const int wave = tid >> 5;   // 0..7
    const int lo   = lane & 15;
    const int hi   = lane >> 4;  // selects K-pair / M+8 half

    const int s0 = blockIdx.x * BN;
    const int nb = blockIdx.y;
    const float* __restrict__ xb = x + (size_t)nb * C_CH * HW;

    const int waveM = wave * 32;

    v8f acc[2][4] = {};  // 2 M-tiles x 4 N-tiles of 16x16 fp32

    for (int k0 = 0; k0 < C_CH; k0 += BK) {
        __syncthreads();  // previous block's LDS reads complete

        // ---- stage x[nb, k0:k0+32, s0:s0+64] -> LDS, K-pair interleaved
        {
            const int p  = tid >> 4;          // k-pair 0..15
            const int sv = (tid & 15) << 2;   // col 0..60 step 4
            const float* r0 = xb + (size_t)(k0 + 2 * p) * HW + s0 + sv;
            const float4 a = *(const float4*)r0;         // row 2p
            const float4 b = *(const float4*)(r0 + HW);  // row 2p+1
            float* dst = &Bs[p * PSTR + sv * 2];
            *(float4*)(dst)     = make_float4(a.x, b.x, a.y, b.y);
            *(float4*)(dst + 4) = make_float4(a.z, b.z, a.w, b.w);

            if (k0 + BK < C_CH) {  // uniform branch; EXEC stays all-ones
                __builtin_prefetch(r0 + (size_t)BK * HW, 0, 0);
                __builtin_prefetch(r0 + (size_t)(BK + 1) * HW, 0, 0);
            }
        }
        __syncthreads();

        // ---- 8 K-steps of 16x16x4 fp32 WMMA
        #pragma unroll
        for (int ks = 0; ks < BK / 4; ++ks) {
            const int kp = 2 * ks + hi;          // local k-pair for my half
            const int kg = k0 + 4 * ks + 2 * hi; // global K row

            // B fragment: single b64 -> {B[k][n], B[k+1][n]}
            v2f bfrag[4];
            #pragma unroll
            for (int nt = 0; nt < 4; ++nt)
                bfrag[nt] = *(const v2f*)&Bs[kp * PSTR + (nt * 16 + lo) * 2];

            // A fragment: a = {Wt[m][kg], Wt[m][kg+1]}
            v2f afrag[2];
            #pragma unroll
            for (int mt = 0; mt < 2; ++mt) {
                const int m = waveM + mt * 16 + lo;
                if constexpr (PRET) {
                    // pair-interleaved, coalesced b64 across the half-wave
                    afrag[mt] = *(const v2f*)&Wsrc[(size_t)(kg >> 1) * 512 + m * 2];
                } else {
                    afrag[mt].x = Wsrc[(size_t)kg * C_CH + m];
                    afrag[mt].y = Wsrc[(size_t)(kg + 1) * C_CH + m];
                }
            }

            #pragma unroll
            for (int mt = 0; mt < 2; ++mt) {
                #pragma unroll
                for (int nt = 0; nt < 4; ++nt) {
                    acc[mt][nt] = __builtin_amdgcn_wmma_f32_16x16x4_f32(
                        /*neg_a=*/false, afrag[mt],
                        /*neg_b=*/false, bfrag[nt],
                        /*c_mod=*/(short)0, acc[mt][nt],
                        /*reuse_a=*/false, /*reuse_b=*/false);
                }
            }
        }
    }

    // ---- store: D layout: VGPR r -> row m = base + r + 8*hi, col = lo
    #pragma unroll
    for (int mt = 0; mt < 2; ++mt) {
        #pragma unroll
        for (int nt = 0; nt < 4; ++nt) {
            const int s = s0 + nt * 16 + lo;
            #pragma unroll
            for (int r = 0; r < 8; ++r) {
                const int j = waveM + mt * 16 + 8 * hi + r;
                out[((size_t)nb * C_CH + j) * HW + s] = acc[mt][nt][r];
            }
        }
    }
}

extern "C" void kernel_launch(void* const* d_in, const int* in_sizes, int n_in,
                              void* d_out, int out_size, void* d_ws, size_t ws_size,
                              hipStream_t stream) {
    const float* x  = (const float*)d_in[0];   // [64, 256, 56, 56] fp32
    const float* Wm = (const float*)d_in[1];   // [256, 256] fp32
    float* out = (float*)d_out;

    const int nbatch = in_sizes[0] / (C_CH * HW);  // 64
    dim3 grid(HW / BN, nbatch);                    // (49, 64)
    dim3 block(256);                               // 8 waves (wave32)

    if (ws_size >= (size_t)C_CH * C_CH * sizeof(float)) {
        float* Wp = (float*)d_ws;
        hipLaunchKernelGGL(transpose_w_pairs, dim3(C_CH * C_CH / 256), dim3(256),
                           0, stream, Wm, Wp);
        hipLaunchKernelGGL(feature_rotation_wmma_f32<true>, grid, block, 0,
                           stream, x, Wp, out);
    } else {
        hipLaunchKernelGGL(feature_rotation_wmma_f32<false>, grid, block, 0,
                           stream, x, Wm, out);
    }
}